// SRBF3d_45561013076209
// MI455X (gfx1250) — compile-verified
//
#include <hip/hip_runtime.h>

// SRBF3d on gfx1250 (MI455X, wave32, WMMA).
//
// exp argument computed as a rank-8 bilinear form via V_WMMA_F32_16X16X4_F32
// (two chained WMMAs -> K=8). 16x16 tiles: rows = 16 x-points per wave,
// cols = 16 RBF centers per chunk, 64 chunks of 1024 centers, 4 sets.
//
// Optimizations vs v1:
//  - B features pre-scaled by log2(e) in LDS -> exp via raw v_exp_f32
//    (exp2), no per-element mul; outputs un-scaled once by ln2 at write.
//  - B operand picked by lane-half via DS *address* (no v_cndmask).
//  - Weight products h*bf precomputed into LDS once per block per set
//    (no per-chunk v_mul).

typedef float v2f __attribute__((ext_vector_type(2)));
typedef float v8f __attribute__((ext_vector_type(8)));

#define NCENT 1024
#define BLOCK 256            // 8 waves * 16 rows = 128 rows per block

#define LOG2E 1.44269504088896340736f
#define LN2   0.69314718055994530942f

// LDS layout (64 KB):
//  B records  : lds[m*8 + k]          = log2e * {-d0,-d1,-d2, 2c0d0,2c1d1,2c2d2, -(c.c)_d, 0}
//  W records  : lds[NCENT*8 + m*8 +k] = {h*B0, h*B3, h*B1, h*B4, h*B2, h*B5, h, 0}
//               (h*Bk are log2e-scaled; h itself is NOT scaled)
__device__ __forceinline__ void precompute_set(const float* __restrict__ h,
                                               const float* __restrict__ c,
                                               const float* __restrict__ w,
                                               float* lds, int tid) {
    for (int m = tid; m < NCENT; m += BLOCK) {
        float c0 = c[3 * m + 0], c1 = c[3 * m + 1], c2 = c[3 * m + 2];
        float w0 = w[3 * m + 0], w1 = w[3 * m + 1], w2 = w[3 * m + 2];
        float d0 = w0 * w0, d1 = w1 * w1, d2 = w2 * w2;
        float b0 = -LOG2E * d0;
        float b1 = -LOG2E * d1;
        float b2 = -LOG2E * d2;
        float b3 = 2.0f * LOG2E * c0 * d0;
        float b4 = 2.0f * LOG2E * c1 * d1;
        float b5 = 2.0f * LOG2E * c2 * d2;
        float b6 = -LOG2E * (c0 * c0 * d0 + c1 * c1 * d1 + c2 * c2 * d2);
        float hm = h[m];
        float4* pb = (float4*)(lds + m * 8);
        pb[0] = make_float4(b0, b1, b2, b3);
        pb[1] = make_float4(b4, b5, b6, 0.0f);
        float4* pw = (float4*)(lds + NCENT * 8 + m * 8);
        pw[0] = make_float4(hm * b0, hm * b3, hm * b1, hm * b4);
        pw[1] = make_float4(hm * b2, hm * b5, hm, 0.0f);
    }
}

// MODE 0: set1 -> ux, uy, uz (out offsets 0,N,2N)
// MODE 1..3: set{2,3,4} -> {P,Px},{Q,Qy},{R,Rz} at offsets (1+2*MODE)N,(2+2*MODE)N
template <int MODE>
__device__ __forceinline__ void run_set(const float* lds, v2f a1, v2f a2,
                                        int laneHi, int lane15,
                                        float xv, float yv, float zv,
                                        int rowBase, float* __restrict__ out,
                                        int N) {
    v8f acc0 = {}, acc1 = {}, acc2 = {}, acc3 = {}, acc4 = {}, acc5 = {};

    // lane-half folded into the B-operand address: no cndmask in the loop
    const float* bbase = lds + lane15 * 8 + laneHi * 2;
    const float* wbase = lds + NCENT * 8 + lane15 * 8;

    for (int chunk = 0; chunk < NCENT / 16; ++chunk) {
        const float* bp = bbase + chunk * 16 * 8;
        v2f b1 = *(const v2f*)(bp);       // lanes<16: (B0,B1)  lanes>=16: (B2,B3)
        v2f b2 = *(const v2f*)(bp + 4);   // lanes<16: (B4,B5)  lanes>=16: (B6,0)

        v8f t = {};
        t = __builtin_amdgcn_wmma_f32_16x16x4_f32(false, a1, false, b1,
                                                  (short)0, t, false, false);
        t = __builtin_amdgcn_wmma_f32_16x16x4_f32(false, a2, false, b2,
                                                  (short)0, t, false, false);

        // t = log2e * exponent  ->  e = exp2(t)
        v8f e;
#pragma unroll
        for (int v = 0; v < 8; ++v) e[v] = __builtin_amdgcn_exp2f(t[v]);

        const float* wp = wbase + chunk * 16 * 8;
        if (MODE == 0) {
            float4 wa = *(const float4*)(wp);      // hB0, hB3, hB1, hB4
            v2f    wb = *(const v2f*)(wp + 4);     // hB2, hB5
#pragma unroll
            for (int v = 0; v < 8; ++v) {
                acc0[v] += e[v] * wa.x;
                acc1[v] += e[v] * wa.y;
                acc2[v] += e[v] * wa.z;
                acc3[v] += e[v] * wa.w;
                acc4[v] += e[v] * wb.x;
                acc5[v] += e[v] * wb.y;
            }
        } else {
            v2f   wd = *(const v2f*)(wp + 2 * (MODE - 1));  // hB(a), hB(a+3)
            float hm = wp[6];
#pragma unroll
            for (int v = 0; v < 8; ++v) {
                acc0[v] += e[v] * hm;
                acc1[v] += e[v] * wd.x;
                acc2[v] += e[v] * wd.y;
            }
        }
    }

    // tree-reduce columns across each 16-lane half
#pragma unroll
    for (int mask = 1; mask < 16; mask <<= 1) {
#pragma unroll
        for (int v = 0; v < 8; ++v) {
            acc0[v] += __shfl_xor(acc0[v], mask, 32);
            acc1[v] += __shfl_xor(acc1[v], mask, 32);
            acc2[v] += __shfl_xor(acc2[v], mask, 32);
            if (MODE == 0) {
                acc3[v] += __shfl_xor(acc3[v], mask, 32);
                acc4[v] += __shfl_xor(acc4[v], mask, 32);
                acc5[v] += __shfl_xor(acc5[v], mask, 32);
            }
        }
    }

    // C/D layout: VGPR v holds rows v (lanes 0-15) and v+8 (lanes 16-31).
    // Lanes 0 and 16 hold full sums; fetch row coords via shuffle.
    // Derivative sums carry a log2e factor -> multiply by ln2 once here.
#pragma unroll
    for (int v = 0; v < 8; ++v) {
        int srcLane = laneHi * 8 + v;           // lanes 0..15 hold rows 0..15
        float xr = __shfl(xv, srcLane, 32);
        float yr = __shfl(yv, srcLane, 32);
        float zr = __shfl(zv, srcLane, 32);
        int row = rowBase + laneHi * 8 + v;
        if (lane15 == 0) {
            if (MODE == 0) {
                out[0 * N + row] = (2.0f * xr * acc0[v] + acc1[v]) * LN2;  // ux
                out[1 * N + row] = (2.0f * yr * acc2[v] + acc3[v]) * LN2;  // uy
                out[2 * N + row] = (2.0f * zr * acc4[v] + acc5[v]) * LN2;  // uz
            } else {
                float cr = (MODE == 1) ? xr : (MODE == 2) ? yr : zr;
                out[(1 + 2 * MODE) * N + row] = acc0[v];                           // P/Q/R
                out[(2 + 2 * MODE) * N + row] = (2.0f * cr * acc1[v] + acc2[v]) * LN2; // Px/Qy/Rz
            }
        }
    }
}

__global__ __launch_bounds__(BLOCK) void srbf3d_wmma_kernel(
    const float* __restrict__ x,
    const float* __restrict__ h1, const float* __restrict__ c1, const float* __restrict__ w1,
    const float* __restrict__ h2, const float* __restrict__ c2, const float* __restrict__ w2,
    const float* __restrict__ h3, const float* __restrict__ c3, const float* __restrict__ w3,
    const float* __restrict__ h4, const float* __restrict__ c4, const float* __restrict__ w4,
    float* __restrict__ out, int N) {
    __shared__ float lds[NCENT * 16];   // 64 KB

    const int tid = threadIdx.x;
    const int lane = tid & 31;
    const int wave = tid >> 5;
    const int laneHi = lane >> 4;      // 0 or 1
    const int lane15 = lane & 15;
    const int rowBase = blockIdx.x * 128 + wave * 16;
    const int myRow = rowBase + lane15;

    float xv = x[3 * myRow + 0];
    float yv = x[3 * myRow + 1];
    float zv = x[3 * myRow + 2];

    // A operand, 16x4 f32: lanes 0-15 hold K={0,1}, lanes 16-31 K={2,3}.
    // Features: [x^2, y^2, z^2, x] (WMMA1) and [y, z, 1, 0] (WMMA2).
    v2f a1 = laneHi ? v2f{zv * zv, xv} : v2f{xv * xv, yv * yv};
    v2f a2 = laneHi ? v2f{1.0f, 0.0f}  : v2f{yv, zv};

    precompute_set(h1, c1, w1, lds, tid);
    __syncthreads();
    run_set<0>(lds, a1, a2, laneHi, lane15, xv, yv, zv, rowBase, out, N);
    __syncthreads();

    precompute_set(h2, c2, w2, lds, tid);
    __syncthreads();
    run_set<1>(lds, a1, a2, laneHi, lane15, xv, yv, zv, rowBase, out, N);
    __syncthreads();

    precompute_set(h3, c3, w3, lds, tid);
    __syncthreads();
    run_set<2>(lds, a1, a2, laneHi, lane15, xv, yv, zv, rowBase, out, N);
    __syncthreads();

    precompute_set(h4, c4, w4, lds, tid);
    __syncthreads();
    run_set<3>(lds, a1, a2, laneHi, lane15, xv, yv, zv, rowBase, out, N);
}

extern "C" void kernel_launch(void* const* d_in, const int* in_sizes, int n_in,
                              void* d_out, int out_size, void* d_ws, size_t ws_size,
                              hipStream_t stream) {
    const float* x  = (const float*)d_in[0];
    const float* h1 = (const float*)d_in[1];
    const float* c1 = (const float*)d_in[2];
    const float* w1 = (const float*)d_in[3];
    const float* h2 = (const float*)d_in[4];
    const float* c2 = (const float*)d_in[5];
    const float* w2 = (const float*)d_in[6];
    const float* h3 = (const float*)d_in[7];
    const float* c3 = (const float*)d_in[8];
    const float* w3 = (const float*)d_in[9];
    const float* h4 = (const float*)d_in[10];
    const float* c4 = (const float*)d_in[11];
    const float* w4 = (const float*)d_in[12];
    float* out = (float*)d_out;

    int N = in_sizes[0] / 3;             // 32768
    dim3 grid(N / 128), block(BLOCK);
    hipLaunchKernelGGL(srbf3d_wmma_kernel, grid, block, 0, stream,
                       x, h1, c1, w1, h2, c2, w2, h3, c3, w3, h4, c4, w4,
                       out, N);
}